// SelectiveScan_19885698581141
// MI455X (gfx1250) — compile-verified
//
#include <hip/hip_runtime.h>
#include <hip/hip_bf16.h>

// ---------------------------------------------------------------------------
// Mamba selective-scan block for MI455X (gfx1250, wave32).
// GEMMs use V_WMMA_F32_16X16X4_F32 (exact fp32 matrix path).
// A tiles: double-buffered LDS, filled by GLOBAL_LOAD_ASYNC_TO_LDS_B128
//          (ASYNCcnt DMA) overlapped with compute of the previous tile.
// B tiles: pipelined through registers, scattered into a k-pair-interleaved
//          LDS layout so every WMMA fragment is one contiguous B64 DS load.
// Dims: D_MODEL=1024, D_STATE=16, D_CONV=4, D_INNER=2048, BATCH=2, SEQ=2048.
// ---------------------------------------------------------------------------

typedef __attribute__((ext_vector_type(2))) float v2f;
typedef __attribute__((ext_vector_type(4))) float v4f;
typedef __attribute__((ext_vector_type(8))) float v8f;

#define D_MODEL 1024
#define D_STATE 16
#define D_CONV  4
#define D_INNER 2048
#define BATCH   2
#define SEQ     2048
#define NROWS   (BATCH * SEQ)          // 4096 rows of activations

enum { EPI_NONE = 0, EPI_SOFTPLUS = 1 };

__device__ __forceinline__ float silu_f(float x) {
    return x / (1.0f + expf(-x));
}

// Async DMA: 16 global bytes -> LDS, tracked by ASYNCcnt (gfx1250).
__device__ __forceinline__ void async_copy_b128(unsigned lds_byte_off,
                                                const float* gsrc) {
    asm volatile("global_load_async_to_lds_b128 %0, %1, off"
                 :: "v"(lds_byte_off), "v"(gsrc)
                 : "memory");
}
__device__ __forceinline__ void wait_async0() {
    asm volatile("s_wait_asynccnt 0" ::: "memory");
}

// ---------------------------------------------------------------------------
// Templated fp32 WMMA GEMM:  C[M,N] = A[M,K] @ B[K,N] + bias[N]  (+epilogue)
//   WM x WN waves per block; each wave computes a 16 x (16*TN) tile.
// WMMA fragment layout (V_WMMA_F32_16X16X4_F32, wave32):
//   A (16x4): lanes 0-15 -> M=lane, {v0,v1}={K0,K1}; lanes 16-31 -> {K2,K3}
//   B (4x16): lanes 0-15 -> N=lane, {v0,v1}={K0,K1}; lanes 16-31 -> {K2,K3}
//   C (16x16): vgpr v, lanes 0-15 -> M=v, N=lane; lanes 16-31 -> M=v+8
// LDS layouts (each fragment = one contiguous 64-bit DS load):
//   sA[buf][m][k]          row stride KT+4=20 dwords (bank-clean, 16B rows)
//   sBp[k/2][2n + (k&1)]   row stride 2*BN+32 dwords (halves in disjoint banks)
// ---------------------------------------------------------------------------
template <int WM, int WN, int TN, int EPI>
__global__ __launch_bounds__(32 * WM * WN)
void gemm_wmma_f32(const float* __restrict__ A, int lda,
                   const float* __restrict__ B, int ldb,
                   const float* __restrict__ bias,
                   float* __restrict__ C, int ldc,
                   int K)
{
    constexpr int KT   = 16;
    constexpr int BM   = 16 * WM;
    constexpr int BN   = 16 * TN * WN;
    constexpr int LDAS = KT + 4;              // 20 dwords per A row
    constexpr int LDBP = 2 * BN + 32;         // paired-B row stride (dwords)
    constexpr int NT   = 32 * WM * WN;
    constexpr int ACHN = BM * 4;              // total A B128 chunks per tile
    constexpr int ACH  = (ACHN + NT - 1) / NT;
    constexpr int BCHN = KT * (BN / 4);       // total B v4f chunks per tile
    constexpr int BCH  = (BCHN + NT - 1) / NT;

    __shared__ __align__(16) float sA[2][BM][LDAS];
    __shared__ __align__(16) float sBp[KT / 2][LDBP];

    const int tid      = threadIdx.x;
    const int lane     = tid & 31;
    const int wave     = tid >> 5;
    const int laneHalf = lane & 15;
    const int hi       = lane >> 4;          // 0: K0/K1 half, 1: K2/K3 half
    const int mBase    = (wave % WM) * 16;
    const int nBase    = (wave / WM) * (16 * TN);
    const long gM0     = (long)blockIdx.x * BM;
    const long gN0     = (long)blockIdx.y * BN;

    auto stageA = [&](int buf, int kt) {
#pragma unroll
        for (int c = 0; c < ACH; ++c) {
            const int idx = tid + c * NT;
            if (ACHN % NT == 0 || idx < ACHN) {
                const int m = idx >> 2;
                const int j = idx & 3;
                async_copy_b128((unsigned)(size_t)&sA[buf][m][4 * j],
                                A + (gM0 + m) * (long)lda + kt + 4 * j);
            }
        }
    };
    auto loadB = [&](int kt, v4f* breg) {
#pragma unroll
        for (int c = 0; c < BCH; ++c) {
            const int idx = tid + c * NT;
            if (BCHN % NT == 0 || idx < BCHN)
                breg[c] = *(const v4f*)(B + (long)(kt + idx / (BN / 4)) * ldb +
                                        gN0 + 4 * (idx % (BN / 4)));
        }
    };
    auto storeB = [&](const v4f* breg) {
#pragma unroll
        for (int c = 0; c < BCH; ++c) {
            const int idx = tid + c * NT;
            if (BCHN % NT == 0 || idx < BCHN) {
                const int k = idx / (BN / 4);
                const int j = idx % (BN / 4);
                float* dst = &sBp[k >> 1][8 * j + (k & 1)];
                dst[0] = breg[c].x;  dst[2] = breg[c].y;
                dst[4] = breg[c].z;  dst[6] = breg[c].w;
            }
        }
    };

    v8f acc[TN];
    const v8f vzero = {};
#pragma unroll
    for (int t = 0; t < TN; ++t) acc[t] = vzero;

    // ---- software pipeline: prime tile 0 ------------------------------
    v4f breg[BCH], bnext[BCH];
    stageA(0, 0);
    loadB(0, breg);

    const int ntiles = K / KT;
    int cur = 0;
    for (int i = 0; i < ntiles; ++i) {
        wait_async0();          // my DMAs for sA[cur] complete
        __syncthreads();        // everyone's DMAs done + prev compute done
        storeB(breg);           // scatter B tile i into LDS
        if (i + 1 < ntiles) {   // prefetch tile i+1 while computing tile i
            stageA(cur ^ 1, (i + 1) * KT);
            loadB((i + 1) * KT, bnext);
        }
        __syncthreads();        // sBp stores visible

        // ---- compute tile i: KT/4 WMMA bursts -------------------------
#pragma unroll
        for (int kk = 0; kk < KT; kk += 4) {
            const int ka = kk + 2 * hi;                 // even
            const v2f afrag = *(const v2f*)&sA[cur][mBase + laneHalf][ka];
            v2f bfrag[TN];
#pragma unroll
            for (int t = 0; t < TN; ++t)
                bfrag[t] =
                    *(const v2f*)&sBp[ka >> 1][2 * (nBase + 16 * t + laneHalf)];
#pragma unroll
            for (int t = 0; t < TN; ++t)
                acc[t] = __builtin_amdgcn_wmma_f32_16x16x4_f32(
                    false, afrag, false, bfrag[t], (short)0, acc[t], false, false);
        }

#pragma unroll
        for (int c = 0; c < BCH; ++c) breg[c] = bnext[c];
        cur ^= 1;
    }

    // ---- epilogue: bias (+softplus), store ----------------------------
#pragma unroll
    for (int t = 0; t < TN; ++t) {
        const long col = gN0 + nBase + 16 * t + laneHalf;
        const float bv = bias ? bias[col] : 0.0f;
#pragma unroll
        for (int v = 0; v < 8; ++v) {
            const long row = gM0 + mBase + v + 8 * hi;
            float val = acc[t][v] + bv;
            if (EPI == EPI_SOFTPLUS)
                val = (val > 20.0f) ? val : log1pf(expf(val));
            C[row * (long)ldc + col] = val;
        }
    }
}

// ---------------------------------------------------------------------------
// Depthwise causal conv (width 4) + bias + silu.
//   xz : [NROWS][2*D_INNER] (first half is pre-conv xh)
//   xh : [NROWS][D_INNER]   (activated output)
// ---------------------------------------------------------------------------
__global__ __launch_bounds__(256)
void conv_silu_kernel(const float* __restrict__ xz,
                      const float* __restrict__ conv_w,   // [D_INNER][4]
                      const float* __restrict__ conv_b,   // [D_INNER]
                      float* __restrict__ xh)
{
    const long idx = (long)blockIdx.x * 256 + threadIdx.x;  // over NROWS*D_INNER
    const int  d   = (int)(idx % D_INNER);
    const long row = idx / D_INNER;                          // b*SEQ + t
    const int  t   = (int)(row % SEQ);

    float acc = conv_b[d];
#pragma unroll
    for (int j = 0; j < D_CONV; ++j) {
        const int tt = t - (D_CONV - 1) + j;
        if (tt >= 0)
            acc += conv_w[d * D_CONV + j] *
                   xz[(row - (D_CONV - 1) + j) * (2 * D_INNER) + d];
    }
    xh[row * D_INNER + d] = silu_f(acc);
}

// ---------------------------------------------------------------------------
// Selective scan, fused with y = (ys + xh*D) * silu(z).
// One wave handles 2 channels; 16 lanes per channel = 16 states.
// Cross-state sum via __shfl_xor butterfly of width 16.
// ---------------------------------------------------------------------------
__global__ __launch_bounds__(256)
void scan_kernel(const float* __restrict__ delta,   // [NROWS][D_INNER]
                 const float* __restrict__ xdbl,    // [NROWS][2*D_STATE]
                 const float* __restrict__ xh,      // [NROWS][D_INNER]
                 const float* __restrict__ xz,      // [NROWS][2*D_INNER] (z half)
                 const float* __restrict__ A_log,   // [D_INNER][D_STATE]
                 const float* __restrict__ Dvec,    // [D_INNER]
                 float* __restrict__ y)             // [NROWS][D_INNER]
{
    const int tid      = threadIdx.x;
    const int lane     = tid & 31;
    const int wave     = tid >> 5;
    const int laneHalf = lane & 15;          // state index n
    const int hi       = lane >> 4;          // which channel within the pair
    const int gw       = blockIdx.x * 8 + wave;
    const int c        = gw * 2 + hi;        // channel 0..D_INNER-1
    const int b        = blockIdx.y;

    const float Aval = -expf(A_log[c * D_STATE + laneHalf]);
    const float Dval = Dvec[c];

    float h = 0.0f;
    long row = (long)b * SEQ;
    for (int t = 0; t < SEQ; ++t, ++row) {
        const float dt = delta[row * D_INNER + c];
        const float u  = xh[row * D_INNER + c];
        const float Bv = xdbl[row * (2 * D_STATE) + D_STATE + laneHalf];

        h = h * expf(dt * Aval) + dt * Bv * u;

        float s = h;
#pragma unroll
        for (int off = 8; off > 0; off >>= 1)
            s += __shfl_xor(s, off, 16);

        if (laneHalf == 0) {
            const float zv = xz[row * (2 * D_INNER) + D_INNER + c];
            y[row * D_INNER + c] = (s + u * Dval) * silu_f(zv);
        }
    }
}

// ---------------------------------------------------------------------------
// Launcher
// ---------------------------------------------------------------------------
extern "C" void kernel_launch(void* const* d_in, const int* in_sizes, int n_in,
                              void* d_out, int out_size, void* d_ws, size_t ws_size,
                              hipStream_t stream) {
    const float* x      = (const float*)d_in[0];
    const float* W_in   = (const float*)d_in[1];
    const float* b_in   = (const float*)d_in[2];
    const float* conv_w = (const float*)d_in[3];
    const float* conv_b = (const float*)d_in[4];
    const float* W_x    = (const float*)d_in[5];
    const float* b_x    = (const float*)d_in[6];
    const float* W_dt   = (const float*)d_in[7];
    const float* b_dt   = (const float*)d_in[8];
    const float* A_log  = (const float*)d_in[9];
    const float* Dvec   = (const float*)d_in[10];
    const float* W_out  = (const float*)d_in[11];
    const float* b_out  = (const float*)d_in[12];
    float* out = (float*)d_out;

    float* ws = (float*)d_ws;
    float* xz    = ws;                                   // [4096][4096]
    float* xh    = xz    + (long)NROWS * (2 * D_INNER);  // [4096][2048]
    float* xdbl  = xh    + (long)NROWS * D_INNER;        // [4096][32]
    float* delta = xdbl  + (long)NROWS * (2 * D_STATE);  // [4096][2048]
    float* ybuf  = delta + (long)NROWS * D_INNER;        // [4096][2048]

    // 1) xz = x @ W_in + b_in          (M=4096, N=4096, K=1024)
    gemm_wmma_f32<4, 2, 8, EPI_NONE><<<dim3(NROWS / 64, (2 * D_INNER) / 256), 256, 0, stream>>>(
        x, D_MODEL, W_in, 2 * D_INNER, b_in, xz, 2 * D_INNER, D_MODEL);

    // 2) xh = silu(depthwise_conv(xz[:, :D_INNER]) + conv_b)
    conv_silu_kernel<<<(long)NROWS * D_INNER / 256, 256, 0, stream>>>(
        xz, conv_w, conv_b, xh);

    // 3) x_dbl = xh @ W_x + b_x        (M=4096, N=32, K=2048)
    gemm_wmma_f32<8, 1, 2, EPI_NONE><<<dim3(NROWS / 128, 1), 256, 0, stream>>>(
        xh, D_INNER, W_x, 2 * D_STATE, b_x, xdbl, 2 * D_STATE, D_INNER);

    // 4) delta = softplus(xh @ W_dt + b_dt)   (M=4096, N=2048, K=2048)
    gemm_wmma_f32<4, 2, 8, EPI_SOFTPLUS><<<dim3(NROWS / 64, D_INNER / 256), 256, 0, stream>>>(
        xh, D_INNER, W_dt, D_INNER, b_dt, delta, D_INNER, D_INNER);

    // 5) selective scan + fused (ys + xh*D) * silu(z)
    scan_kernel<<<dim3(D_INNER / 16, BATCH), 256, 0, stream>>>(
        delta, xdbl, xh, xz, A_log, Dvec, ybuf);

    // 6) out = y @ W_out + b_out       (M=4096, N=1024, K=2048)
    gemm_wmma_f32<4, 2, 8, EPI_NONE><<<dim3(NROWS / 64, D_MODEL / 256), 256, 0, stream>>>(
        ybuf, D_INNER, W_out, D_MODEL, b_out, out, D_MODEL, D_INNER);
}